// GATv2Conv_40673340293246
// MI455X (gfx1250) — compile-verified
//
#include <hip/hip_runtime.h>
#include <hip/hip_bf16.h>

typedef __attribute__((ext_vector_type(2))) float v2f;
typedef __attribute__((ext_vector_type(8))) float v8f;

#define N_NODES 100000
#define N_EDGES 300000
#define D 256
#define NEG_SLOPE 0.2f

// ---------------- init: zero out / ssum, smax keys to -inf ----------------
__global__ void k_init(float* __restrict__ out, float* __restrict__ ssum,
                       int* __restrict__ smaxKey) {
  long long i = (long long)blockIdx.x * blockDim.x + threadIdx.x;
  const long long total = (long long)N_NODES * D;
  if (i < total) out[i] = 0.0f;
  if (i < N_NODES) { ssum[i] = 0.0f; smaxKey[i] = (int)0x80000000; }
}

// ---------------- projections via WMMA fp32 16x16x4 ----------------
// One workgroup per 32-row strip of feat (100000 = 32*3125, exact -> EXEC all-1s).
// 8 waves: waves 0-3 -> el (W_src), waves 4-7 -> er (W_dst); each wave owns a
// 64-col slab (4 N-tiles) and computes BOTH 16-row M-tiles of the strip,
// reusing each B fragment for 2 WMMAs (halves W L2 traffic vs 16-row strips).
__global__ __launch_bounds__(256) void k_proj(
    const float* __restrict__ feat,
    const float* __restrict__ Wsrc, const float* __restrict__ bsrc,
    const float* __restrict__ Wdst, const float* __restrict__ bdst,
    float* __restrict__ el, float* __restrict__ er)
{
  __shared__ float A[32][260];   // 32x256 feat tile, padded row (1040B = 65*16, 16B aligned)
  const int tid  = threadIdx.x;
  const int row0 = blockIdx.x * 32;

  // cooperative fill: 256 threads x 32 floats (8 float4 each)
  {
    const int r   = tid >> 3;              // 0..31
    const int seg = tid & 7;               // 8 segments of 32 floats
    const float4* g = (const float4*)(feat + (long long)(row0 + r) * D + seg * 32);
    float4* s = (float4*)(&A[r][seg * 32]);
#pragma unroll
    for (int j = 0; j < 8; ++j) s[j] = g[j];
  }
  __syncthreads();

  const int lane = tid & 31;
  const int wave = tid >> 5;      // 0..7
  const int half = wave >> 2;     // 0 -> el, 1 -> er
  const int quad = wave & 3;      // which 64-col slab
  const int mn   = lane & 15;     // row-in-tile (A) / col-in-tile (B,C,D)
  const int hi   = lane >> 4;     // lane-half selector per ISA layouts

  const float* __restrict__ W    = half ? Wdst : Wsrc;
  const float* __restrict__ bias = half ? bdst : bsrc;
  float* __restrict__ outp       = half ? er : el;

  v8f acc[2][4];
#pragma unroll
  for (int mt = 0; mt < 2; ++mt)
#pragma unroll
    for (int t = 0; t < 4; ++t)
      acc[mt][t] = (v8f){0.f,0.f,0.f,0.f,0.f,0.f,0.f,0.f};

  const int ncol0 = quad * 64 + mn;

  for (int k0 = 0; k0 < D; k0 += 4) {
    const int kb = k0 + 2 * hi;           // ISA 16x4 f32 A layout: VGPR0=K0/K2, VGPR1=K1/K3
    v2f a0, a1;
    a0.x = A[mn][kb];       a0.y = A[mn][kb + 1];        // M-tile 0 (rows 0-15)
    a1.x = A[mn + 16][kb];  a1.y = A[mn + 16][kb + 1];   // M-tile 1 (rows 16-31)
    const float* wr0 = W + (long long)kb * D;
    const float* wr1 = W + (long long)(kb + 1) * D;
#pragma unroll
    for (int t = 0; t < 4; ++t) {
      const int n = ncol0 + t * 16;
      v2f b;
      b.x = wr0[n];                       // B 4x16: row striped across lanes per VGPR
      b.y = wr1[n];
      acc[0][t] = __builtin_amdgcn_wmma_f32_16x16x4_f32(
          false, a0, false, b, (short)0, acc[0][t], false, false);
      acc[1][t] = __builtin_amdgcn_wmma_f32_16x16x4_f32(
          false, a1, false, b, (short)0, acc[1][t], false, false);
    }
  }

  // epilogue: bias add + store (C/D layout: VGPR v -> row v + 8*hi, col = lane&15)
#pragma unroll
  for (int mt = 0; mt < 2; ++mt) {
#pragma unroll
    for (int t = 0; t < 4; ++t) {
      const int col = ncol0 + t * 16;
      const float bv = bias[col];
#pragma unroll
      for (int v = 0; v < 8; ++v) {
        const int r = mt * 16 + v + 8 * hi;
        outp[(long long)(row0 + r) * D + col] = acc[mt][t][v] + bv;
      }
    }
  }
}

// ---------------- per-edge score + segment max (ordered-int atomicMax) -----
__global__ __launch_bounds__(256) void k_score(
    const float* __restrict__ el, const float* __restrict__ er,
    const float* __restrict__ attn,
    const int* __restrict__ src, const int* __restrict__ dst,
    float* __restrict__ score, int* __restrict__ smaxKey)
{
  const int wave = threadIdx.x >> 5;
  const int lane = threadIdx.x & 31;
  const long long e = (long long)blockIdx.x * 8 + wave;
  if (e >= N_EDGES) return;
  const int s = src[e], d = dst[e];
  const float4* a4 = (const float4*)(el + (long long)s * D);
  const float4* b4 = (const float4*)(er + (long long)d * D);
  const float4* t4 = (const float4*)attn;
  float sum = 0.f;
#pragma unroll
  for (int h = 0; h < 2; ++h) {
    const float4 a = a4[lane + 32 * h];
    const float4 b = b4[lane + 32 * h];
    const float4 t = t4[lane + 32 * h];
    float x;
    x = a.x + b.x; sum += (x > 0.f ? x : NEG_SLOPE * x) * t.x;
    x = a.y + b.y; sum += (x > 0.f ? x : NEG_SLOPE * x) * t.y;
    x = a.z + b.z; sum += (x > 0.f ? x : NEG_SLOPE * x) * t.z;
    x = a.w + b.w; sum += (x > 0.f ? x : NEG_SLOPE * x) * t.w;
  }
#pragma unroll
  for (int off = 16; off >= 1; off >>= 1) sum += __shfl_xor(sum, off, 32);
  if (lane == 0) {
    score[e] = sum;
    int key = __float_as_int(sum);
    key = key >= 0 ? key : (key ^ 0x7FFFFFFF);   // monotone int encoding
    atomicMax(smaxKey + d, key);
  }
}

// ---------------- exp(score - smax[dst]) + segment sum ----------------
__global__ void k_exp(const int* __restrict__ dst,
                      float* __restrict__ score /* in: score, out: ex */,
                      const int* __restrict__ smaxKey,
                      float* __restrict__ ssum) {
  const long long e = (long long)blockIdx.x * blockDim.x + threadIdx.x;
  if (e >= N_EDGES) return;
  const int d = dst[e];
  int key = smaxKey[d];
  const float m = __int_as_float(key >= 0 ? key : (key ^ 0x7FFFFFFF));
  const float x = expf(score[e] - m);
  score[e] = x;
  atomicAdd(ssum + d, x);
}

// ---------------- weighted scatter-sum: out[dst] += el[src] * a -----------
__global__ __launch_bounds__(256) void k_agg(
    const float* __restrict__ el, const float* __restrict__ ex,
    const float* __restrict__ ssum,
    const int* __restrict__ src, const int* __restrict__ dst,
    float* __restrict__ out)
{
  const int wave = threadIdx.x >> 5;
  const int lane = threadIdx.x & 31;
  const long long e = (long long)blockIdx.x * 8 + wave;
  if (e >= N_EDGES) return;
  const int s = src[e], d = dst[e];
  const float a = ex[e] / ssum[d];
  const float4* e4 = (const float4*)(el + (long long)s * D);
  float* o = out + (long long)d * D;
#pragma unroll
  for (int h = 0; h < 2; ++h) {
    const float4 v = e4[lane + 32 * h];
    const int base = (lane + 32 * h) * 4;
    atomicAdd(o + base + 0, v.x * a);
    atomicAdd(o + base + 1, v.y * a);
    atomicAdd(o + base + 2, v.z * a);
    atomicAdd(o + base + 3, v.w * a);
  }
}

extern "C" void kernel_launch(void* const* d_in, const int* in_sizes, int n_in,
                              void* d_out, int out_size, void* d_ws, size_t ws_size,
                              hipStream_t stream) {
  const float* feat = (const float*)d_in[0];
  const float* Wsrc = (const float*)d_in[1];
  const float* bsrc = (const float*)d_in[2];
  const float* Wdst = (const float*)d_in[3];
  const float* bdst = (const float*)d_in[4];
  const float* attn = (const float*)d_in[5];
  const int*   src  = (const int*)d_in[6];
  const int*   dst  = (const int*)d_in[7];
  float* out = (float*)d_out;

  // workspace layout
  char* ws = (char*)d_ws;
  float* el      = (float*)ws;                                    // N*D
  float* er      = el + (long long)N_NODES * D;                   // N*D
  float* score   = er + (long long)N_NODES * D;                   // E (reused as ex)
  int*   smaxKey = (int*)(score + N_EDGES);                       // N
  float* ssum    = (float*)(smaxKey + N_NODES);                   // N

  // 1) init
  {
    const long long total = (long long)N_NODES * D;
    const int blocks = (int)((total + 255) / 256);
    k_init<<<blocks, 256, 0, stream>>>(out, ssum, smaxKey);
  }
  // 2) projections (WMMA, 32-row strips)
  k_proj<<<N_NODES / 32, 256, 0, stream>>>(feat, Wsrc, bsrc, Wdst, bdst, el, er);
  // 3) edge scores + segment max
  k_score<<<(N_EDGES + 7) / 8, 256, 0, stream>>>(el, er, attn, src, dst, score, smaxKey);
  // 4) exp + segment sum
  k_exp<<<(N_EDGES + 255) / 256, 256, 0, stream>>>(dst, score, smaxKey, ssum);
  // 5) weighted aggregation
  k_agg<<<(N_EDGES + 7) / 8, 256, 0, stream>>>(el, score, ssum, src, dst, out);
}